// DotProductPredictionHead_41059887350447
// MI455X (gfx1250) — compile-verified
//
#include <hip/hip_runtime.h>

typedef __attribute__((ext_vector_type(2))) float v2f;
typedef __attribute__((ext_vector_type(8))) float v8f;

#define BATCH 16384
#define DM 64
#define NCAND 200
#define ROWS_PER_BLOCK 16
#define BLOCK_THREADS 256

// Fused: h = relu(x @ W^T + b_lin) via V_WMMA_F32_16X16X4_F32, then
// logits[b,c] = dot(h[b], emb_table[cand[b,c]]) + bias[cand[b,c]].
__global__ __launch_bounds__(BLOCK_THREADS) void dot_pred_head_kernel(
    const float* __restrict__ x,          // [BATCH, DM]
    const int*   __restrict__ candidates, // [BATCH, NCAND]
    const float* __restrict__ W,          // [DM, DM]   (h = x @ W^T)
    const float* __restrict__ b_lin,      // [DM]
    const float* __restrict__ emb_table,  // [VOCAB, DM]
    const float* __restrict__ bias,       // [VOCAB]
    float*       __restrict__ out)        // [BATCH, NCAND]
{
    __shared__ float h_lds[ROWS_PER_BLOCK][DM];

    const int tid  = threadIdx.x;
    const int lane = tid & 31;
    const int wave = tid >> 5;
    const int row0 = blockIdx.x * ROWS_PER_BLOCK;

    // ---- Stage 1: waves 0..3 each compute one 16x16 tile of h via WMMA ----
    if (wave < 4) {
        const int n0   = wave * 16;      // output-column tile
        const int half = lane >> 4;      // 0: lanes 0-15, 1: lanes 16-31
        const int l15  = lane & 15;
        const int arow = row0 + l15;     // A-matrix row (batch row) for this lane
        const int bcol = n0 + l15;       // B-matrix col == W row for this lane

        v8f acc = {0.f, 0.f, 0.f, 0.f, 0.f, 0.f, 0.f, 0.f};
        #pragma unroll
        for (int k0 = 0; k0 < DM; k0 += 4) {
            // ISA 16x4 f32 A layout: lanes<16 hold K=k0,k0+1; lanes>=16 hold K=k0+2,k0+3
            const int kb = k0 + 2 * half;
            v2f a, b;
            a.x = x[arow * DM + kb];
            a.y = x[arow * DM + kb + 1];
            // B[k][n] = W[n][k]  (because h = x @ W^T); same half/K striping as A
            b.x = W[bcol * DM + kb];
            b.y = W[bcol * DM + kb + 1];
            // 8 args: (neg_a, A, neg_b, B, c_mod, C, reuse_a, reuse_b)
            acc = __builtin_amdgcn_wmma_f32_16x16x4_f32(
                false, a, false, b, (short)0, acc, false, false);
        }
        // C/D layout: VGPR v -> M = v + 8*half, N = l15 (+ n0). Fuse bias + ReLU.
        const float bl = b_lin[n0 + l15];
        #pragma unroll
        for (int v = 0; v < 8; ++v) {
            const int m = v + 8 * half;
            float val = acc[v] + bl;
            h_lds[m][n0 + l15] = val > 0.f ? val : 0.f;
        }
    }
    __syncthreads();

    // ---- Stage 2: gather-bound scoring; one candidate per thread ----
    const int total = ROWS_PER_BLOCK * NCAND;
    for (int idx = tid; idx < total; idx += BLOCK_THREADS) {
        const int r = idx / NCAND;
        const int c = idx - r * NCAND;
        const int brow = row0 + r;
        const int cand = candidates[brow * NCAND + c];

        // Prefetch next iteration's embedding row (global_prefetch_b8).
        const int idx2 = idx + BLOCK_THREADS;
        if (idx2 < total) {
            const int r2 = idx2 / NCAND;
            const int c2 = idx2 - r2 * NCAND;
            const int cand2 = candidates[(row0 + r2) * NCAND + c2];
            __builtin_prefetch(emb_table + (size_t)cand2 * DM, 0, 0);
        }

        const float4* __restrict__ e4 = (const float4*)(emb_table + (size_t)cand * DM);
        const float4* __restrict__ h4 = (const float4*)(&h_lds[r][0]);
        float dot = 0.f;
        #pragma unroll
        for (int j = 0; j < DM / 4; ++j) {
            const float4 hv = h4[j];   // LDS broadcast (uniform per sub-warp group)
            const float4 ev = e4[j];   // global_load_b128, sequential 256B row
            dot = fmaf(hv.x, ev.x, dot);
            dot = fmaf(hv.y, ev.y, dot);
            dot = fmaf(hv.z, ev.z, dot);
            dot = fmaf(hv.w, ev.w, dot);
        }
        out[brow * NCAND + c] = dot + bias[cand];
    }
}

extern "C" void kernel_launch(void* const* d_in, const int* in_sizes, int n_in,
                              void* d_out, int out_size, void* d_ws, size_t ws_size,
                              hipStream_t stream) {
    (void)in_sizes; (void)n_in; (void)out_size; (void)d_ws; (void)ws_size;
    const float* x          = (const float*)d_in[0];
    // d_in[1] = b_seq: unused by the reference computation
    const int*   candidates = (const int*)  d_in[2];
    const float* W          = (const float*)d_in[3];
    const float* b_lin      = (const float*)d_in[4];
    const float* emb_table  = (const float*)d_in[5];
    const float* bias       = (const float*)d_in[6];
    float*       out        = (float*)d_out;

    dim3 grid(BATCH / ROWS_PER_BLOCK);   // 1024 blocks x 16 rows
    dim3 block(BLOCK_THREADS);
    dot_pred_head_kernel<<<grid, block, 0, stream>>>(
        x, candidates, W, b_lin, emb_table, bias, out);
}